// kMeanClusterStep_55714315764173
// MI455X (gfx1250) — compile-verified
//
#include <hip/hip_runtime.h>

// k-means cluster step for MI455X (gfx1250, wave32).
//   locF: (8192,1,128) f32, Ck: (1,512,128) f32
//   out : Ck1 (512*128 f32) ++ nItems (512, written as exact float counts)
//
// fp32 WMMA (V_WMMA_16X16X4_F32) for the distance GEMM, LDS-staged Ck tiles
// (double-buffered via GLOBAL_LOAD_ASYNC_TO_LDS_B128), L2 fp32 atomics scatter.

#define KC 512
#define DD 128
#define NN 8192
#define NT (KC / 16)  // 32 column tiles

typedef __attribute__((ext_vector_type(2))) float v2f;
typedef __attribute__((ext_vector_type(8))) float v8f;
typedef __attribute__((ext_vector_type(4))) int   v4i;

#define AS_GLOBAL __attribute__((address_space(1)))
#define AS_LDS    __attribute__((address_space(3)))

#if __has_builtin(__builtin_amdgcn_global_load_async_to_lds_b128) && \
    __has_builtin(__builtin_amdgcn_s_wait_asynccnt)
#define HAVE_ASYNC 1
#else
#define HAVE_ASYNC 0
#endif

__global__ void kmeans_zero_kernel(float* __restrict__ out, int n) {
    int i = blockIdx.x * blockDim.x + threadIdx.x;
    if (i < n) out[i] = 0.0f;
}

// Copy one 16x128 f32 Ck tile (8 KB) into LDS: 512 16B chunks over 256 threads.
__device__ __forceinline__ void stage_tile_async(float* dst, const float* src, int tid) {
#if HAVE_ASYNC
    __builtin_amdgcn_global_load_async_to_lds_b128(
        (AS_GLOBAL v4i*)(AS_GLOBAL void*)(src + tid * 4),
        (AS_LDS    v4i*)(AS_LDS    void*)(dst + tid * 4), 0, 0);
    __builtin_amdgcn_global_load_async_to_lds_b128(
        (AS_GLOBAL v4i*)(AS_GLOBAL void*)(src + (tid + 256) * 4),
        (AS_LDS    v4i*)(AS_LDS    void*)(dst + (tid + 256) * 4), 0, 0);
#endif
}

__global__ __launch_bounds__(256)
void kmeans_step_kernel(const float* __restrict__ X,   // [NN][DD]
                        const float* __restrict__ C,   // [KC][DD]
                        float* __restrict__ outCk,     // [KC][DD]
                        float* __restrict__ outCnt) {  // [KC]
    __shared__ float bs[2][16 * DD];  // double-buffered Ck tile (2 x 8 KB)
    __shared__ float cn[KC];          // centroid squared norms
    __shared__ int   widx[8][16];     // per-wave argmin of its 16 rows

    const int tid  = threadIdx.x;
    const int lane = tid & 31;
    const int w    = tid >> 5;        // wave id 0..7
    const int cl   = lane & 15;       // column / row-in-tile selector
    const int koff = (lane >> 4) * 2; // K sub-offset per half-wave (A/B layout)

    // ---- centroid norms: each thread does 2 clusters (Ck is L2-resident) ----
    for (int c = tid; c < KC; c += 256) {
        const float* cp = C + c * DD;
        float s = 0.0f;
#pragma unroll 8
        for (int d = 0; d < DD; ++d) s = fmaf(cp[d], cp[d], s);
        cn[c] = s;
    }

    const int rowbase = blockIdx.x * 128 + w * 16;  // this wave's 16 rows

    // ---- preload A fragments for all 32 k-steps (read X once) ----
    // A 16x4 f32 layout: lanes 0-15 hold {K=k0,k0+1} of row (lane&15),
    //                    lanes 16-31 hold {K=k0+2,k0+3} of the same rows.
    v2f aFrag[32];
    {
        const float* xr = X + (rowbase + cl) * DD + koff;
#pragma unroll
        for (int s = 0; s < 32; ++s) aFrag[s] = *(const v2f*)(xr + 4 * s);
    }

    // ---- stage tile 0 into LDS buffer 0 ----
#if HAVE_ASYNC
    stage_tile_async(bs[0], C, tid);
    __builtin_amdgcn_s_wait_asynccnt(0);
#else
    {
        const float4* src = (const float4*)C;
        float4 p0 = src[tid], p1 = src[tid + 256];
        float4* dst = (float4*)bs[0];
        dst[tid] = p0; dst[tid + 256] = p1;
    }
#endif
    __syncthreads();

    // Running (min score, argmin) per C/D slot.
    // slot r <-> row rowbase+r (lanes 0-15) / rowbase+8+r (lanes 16-31); col = cl.
    float minv[8];
    int   mini[8];
#pragma unroll
    for (int r = 0; r < 8; ++r) { minv[r] = 3.0e38f; mini[r] = 0; }

    int buf = 0;
#pragma unroll 1
    for (int t = 0; t < NT; ++t) {
        // -- prefetch next tile into the other buffer (overlaps with WMMA) --
#if HAVE_ASYNC
        if (t + 1 < NT) stage_tile_async(bs[buf ^ 1], C + (t + 1) * 16 * DD, tid);
#else
        float4 p0, p1;
        if (t + 1 < NT) {
            const float4* src = (const float4*)(C + (t + 1) * 16 * DD);
            p0 = src[tid]; p1 = src[tid + 256];
        }
#endif

        // -- batch-load all 32 B fragments from LDS, then run the WMMA chain --
        const float* brow = &bs[buf][cl * DD + koff];
        v2f bFrag[32];
#pragma unroll
        for (int s = 0; s < 32; ++s) bFrag[s] = *(const v2f*)(brow + 4 * s);

        v8f acc = {};
#pragma unroll
        for (int s = 0; s < 32; ++s) {
            acc = __builtin_amdgcn_wmma_f32_16x16x4_f32(
                false, aFrag[s], false, bFrag[s], (short)0, acc, false, false);
        }

        const int   ncol = t * 16 + cl;
        const float cnv  = cn[ncol];
#pragma unroll
        for (int r = 0; r < 8; ++r) {
            float sc = cnv - 2.0f * acc[r];   // ||c||^2 - 2 x.c  (argmin-equiv)
            bool lt = sc < minv[r];           // strict < keeps first min index
            mini[r] = lt ? ncol : mini[r];
            minv[r] = lt ? sc : minv[r];
        }

        // -- commit the prefetch, then flip buffers --
#if HAVE_ASYNC
        if (t + 1 < NT) __builtin_amdgcn_s_wait_asynccnt(0);
#else
        if (t + 1 < NT) {
            float4* dst = (float4*)bs[buf ^ 1];
            dst[tid] = p0; dst[tid + 256] = p1;
        }
#endif
        __syncthreads();
        buf ^= 1;
    }

    // ---- per-row min across the 16 lanes holding that row ----
#pragma unroll
    for (int r = 0; r < 8; ++r) {
#pragma unroll
        for (int off = 8; off >= 1; off >>= 1) {
            float ov = __shfl_xor(minv[r], off, 16);
            int   oi = __shfl_xor(mini[r], off, 16);
            bool take = (ov < minv[r]) || (ov == minv[r] && oi < mini[r]);
            minv[r] = take ? ov : minv[r];
            mini[r] = take ? oi : mini[r];
        }
        if (cl == 0) widx[w][r + (lane >> 4) * 8] = mini[r];
    }
    __syncthreads();

    // ---- scatter: segment-sum rows into their centroid accumulator ----
#pragma unroll 1
    for (int rr = 0; rr < 16; ++rr) {
        const int idx = widx[w][rr];
        const float* xr = X + (rowbase + rr) * DD;
#pragma unroll
        for (int j = 0; j < 4; ++j) {
            const int d = lane + 32 * j;
            atomicAdd(&outCk[idx * DD + d], xr[d]);
        }
        if (lane == 0) atomicAdd(&outCnt[idx], 1.0f);
    }
}

extern "C" void kernel_launch(void* const* d_in, const int* in_sizes, int n_in,
                              void* d_out, int out_size, void* d_ws, size_t ws_size,
                              hipStream_t stream) {
    const float* X = (const float*)d_in[0];  // locF (8192,1,128)
    const float* C = (const float*)d_in[1];  // Ck   (1,512,128)
    float* outCk  = (float*)d_out;           // 512*128 sums
    float* outCnt = outCk + KC * DD;         // 512 counts (as exact floats)

    const int total = KC * DD + KC;  // 66048
    kmeans_zero_kernel<<<(total + 255) / 256, 256, 0, stream>>>((float*)d_out, total);

    // 64 blocks x 256 threads; each block covers 128 rows (8 waves x 16 rows)
    kmeans_step_kernel<<<NN / 128, 256, 0, stream>>>(X, C, outCk, outCnt);
}